// ORIGINAL_SSIM_26070451486792
// MI455X (gfx1250) — compile-verified
//
#include <hip/hip_runtime.h>

// ---------------------------------------------------------------------------
// SSIM (16,3,512,512) fp32 — fused separable Gaussian conv via
// V_WMMA_F32_16X16X4_F32 banded GEMMs on gfx1250 (wave32).
// ---------------------------------------------------------------------------

typedef float v2f __attribute__((ext_vector_type(2)));
typedef float v8f __attribute__((ext_vector_type(8)));

#define H_IMG    512
#define W_IMG    512
#define OUT_DIM  502            // 512 - 11 + 1
#define TILE     32             // output tile (rows == cols)
#define NTILES   16             // ceil(502/32)
#define NPLANES  48             // 16 batch * 3 channels
#define NBLOCKS  (NTILES * NTILES * NPLANES)

#define IN_ROWS  48             // 3 row-tiles of 16 (42 needed)
#define IN_STR   44             // 42 needed; stride 44 => conflict-free A reads
#define H_ROWS   48
#define H_STR    44             // +8 rows = 352 dw = 32 banks; +2 rows = 24 banks

// 1D Gaussian, ws=11 sigma=1.5, normalized (matches reference _make_window)
__constant__ float GW[11] = {
    0.00102838f, 0.00759880f, 0.03600083f, 0.10935972f, 0.21300537f,
    0.26601270f,
    0.21300537f, 0.10935972f, 0.03600083f, 0.00759880f, 0.00102838f};

__device__ __forceinline__ float band(int k, int i) {
    int d = k - i;
    return (d >= 0 && d < 11) ? GW[d] : 0.0f;
}

__device__ __forceinline__ v8f wmma4(v2f a, v2f b, v8f c) {
    // D = A(16x4,f32) * B(4x16,f32) + C(16x16,f32)
    return __builtin_amdgcn_wmma_f32_16x16x4_f32(
        false, a, false, b, (short)0, c, false, false);
}

__global__ __launch_bounds__(128) void ssim_tile_kernel(
    const float* __restrict__ img1, const float* __restrict__ img2,
    float* __restrict__ partial) {
    __shared__ float sIn[2][IN_ROWS * IN_STR];     // x1, x2 tiles (+halo)
    __shared__ float sH[5][H_ROWS * H_STR];        // horizontal conv of 5 quantities
    __shared__ float sRed[128];

    const int tid  = threadIdx.x;
    const int lane = tid & 31;
    const int wave = tid >> 5;
    const int lo   = lane & 15;      // M (A rows) / N (B cols) index
    const int hi   = lane >> 4;      // K split across lane halves
    const int kadd = hi * 2;

    const int plane = blockIdx.z;
    const int y0 = blockIdx.y * TILE;
    const int x0 = blockIdx.x * TILE;
    const size_t pbase = (size_t)plane * (H_IMG * W_IMG);

    // ---- Stage 0: load input tiles (clamped at image edge; clamped texels
    //      only ever feed discarded outputs or zero band weights) ----
    for (int i = tid; i < IN_ROWS * IN_STR; i += 128) {
        int r = i / IN_STR;
        int c = i - r * IN_STR;
        int gy = y0 + r; gy = gy < (H_IMG - 1) ? gy : (H_IMG - 1);
        int gx = x0 + c; gx = gx < (W_IMG - 1) ? gx : (W_IMG - 1);
        size_t g = pbase + (size_t)gy * W_IMG + gx;
        sIn[0][i] = img1[g];
        sIn[1][i] = img2[g];
    }

    // Constant banded-Gaussian fragments. Per fragment layout (f32, 16x4 A /
    // 4x16 B): VGPR v, lane L -> K = 4*chunk + v + 2*(L>>4), row/col = L&15.
    // Serves as B for horizontal (Gh[k][n]=g[k-n]) and A for vertical
    // (Gv[m][k]=g[k-m]) — identical per-lane values.
    v2f gb[7];
#pragma unroll
    for (int c = 0; c < 7; ++c) {
        gb[c].x = band(4 * c + kadd + 0, lo);
        gb[c].y = band(4 * c + kadd + 1, lo);
    }

    __syncthreads();

    // ---- Stage 1: horizontal pass. 6 jobs (3 row-tiles x 2 col-tiles),
    //      each job: 5 quantities x 7 K-chunks of WMMA f32 16x16x4 ----
    for (int j = wave; j < 6; j += 4) {
        const int row0 = (j >> 1) * 16;
        const int col0 = (j & 1) * 16;
        v8f acc1 = {0,0,0,0,0,0,0,0};
        v8f acc2 = {0,0,0,0,0,0,0,0};
        v8f acc11 = {0,0,0,0,0,0,0,0};
        v8f acc22 = {0,0,0,0,0,0,0,0};
        v8f acc12 = {0,0,0,0,0,0,0,0};
#pragma unroll
        for (int c = 0; c < 7; ++c) {
            const int base = (row0 + lo) * IN_STR + col0 + 4 * c + kadd;
            v2f a1, a2;
            a1.x = sIn[0][base];     a1.y = sIn[0][base + 1];
            a2.x = sIn[1][base];     a2.y = sIn[1][base + 1];
            acc1  = wmma4(a1,      gb[c], acc1);
            acc2  = wmma4(a2,      gb[c], acc2);
            acc11 = wmma4(a1 * a1, gb[c], acc11);
            acc22 = wmma4(a2 * a2, gb[c], acc22);
            acc12 = wmma4(a1 * a2, gb[c], acc12);
        }
        // C layout: VGPR v, lane L -> row = v + 8*(L>>4), col = L&15
#pragma unroll
        for (int v = 0; v < 8; ++v) {
            const int idx = (row0 + v + 8 * hi) * H_STR + col0 + lo;
            sH[0][idx] = acc1[v];
            sH[1][idx] = acc2[v];
            sH[2][idx] = acc11[v];
            sH[3][idx] = acc22[v];
            sH[4][idx] = acc12[v];
        }
    }

    __syncthreads();

    // ---- Stage 2: vertical pass, one 16x16 output tile per wave (4 tiles).
    //      A = constant Gaussian band, B = H rows from LDS.  SSIM is then
    //      evaluated directly on the 5 accumulator fragments. ----
    const int tY = wave >> 1, tX = wave & 1;
    const int rbase = tY * 16, xb = tX * 16;
    v8f m1 = {0,0,0,0,0,0,0,0};
    v8f m2 = {0,0,0,0,0,0,0,0};
    v8f e11 = {0,0,0,0,0,0,0,0};
    v8f e22 = {0,0,0,0,0,0,0,0};
    v8f e12 = {0,0,0,0,0,0,0,0};
#pragma unroll
    for (int c = 0; c < 7; ++c) {
        const int i0 = (rbase + 4 * c + kadd) * H_STR + xb + lo;
        const int i1 = i0 + H_STR;
        v2f b0, b1, b2, b3, b4;
        b0.x = sH[0][i0]; b0.y = sH[0][i1];
        b1.x = sH[1][i0]; b1.y = sH[1][i1];
        b2.x = sH[2][i0]; b2.y = sH[2][i1];
        b3.x = sH[3][i0]; b3.y = sH[3][i1];
        b4.x = sH[4][i0]; b4.y = sH[4][i1];
        m1  = wmma4(gb[c], b0, m1);
        m2  = wmma4(gb[c], b1, m2);
        e11 = wmma4(gb[c], b2, e11);
        e22 = wmma4(gb[c], b3, e22);
        e12 = wmma4(gb[c], b4, e12);
    }

    const float C1v = 0.0004f;   // (0.01*2)^2
    const float C2v = 0.0036f;   // (0.03*2)^2
    float ssum = 0.0f;
#pragma unroll
    for (int v = 0; v < 8; ++v) {
        const int gy = y0 + rbase + v + 8 * hi;
        const int gx = x0 + xb + lo;
        const float mu1 = m1[v], mu2 = m2[v];
        const float mu1s = mu1 * mu1, mu2s = mu2 * mu2, mu12 = mu1 * mu2;
        const float s1  = e11[v] - mu1s;
        const float s2  = e22[v] - mu2s;
        const float s12 = e12[v] - mu12;
        const float num = (2.0f * mu12 + C1v) * (2.0f * s12 + C2v);
        const float den = (mu1s + mu2s + C1v) * (s1 + s2 + C2v);
        if (gy < OUT_DIM && gx < OUT_DIM) ssum += num / den;
    }

    // ---- Block reduction -> deterministic per-block partial sum ----
    sRed[tid] = ssum;
    __syncthreads();
#pragma unroll
    for (int s = 64; s > 0; s >>= 1) {
        if (tid < s) sRed[tid] += sRed[tid + s];
        __syncthreads();
    }
    if (tid == 0) {
        const int bid = (blockIdx.z * gridDim.y + blockIdx.y) * gridDim.x + blockIdx.x;
        partial[bid] = sRed[0];
    }
}

__global__ __launch_bounds__(256) void ssim_final_kernel(
    const float* __restrict__ partial, float* __restrict__ out, int n) {
    __shared__ float red[256];
    float s = 0.0f;
    for (int i = threadIdx.x; i < n; i += 256) s += partial[i];
    red[threadIdx.x] = s;
    __syncthreads();
#pragma unroll
    for (int t = 128; t > 0; t >>= 1) {
        if (threadIdx.x < t) red[threadIdx.x] += red[threadIdx.x + t];
        __syncthreads();
    }
    if (threadIdx.x == 0)
        out[0] = 1.0f - red[0] / 12096192.0f;   // 48 * 502 * 502
}

extern "C" void kernel_launch(void* const* d_in, const int* in_sizes, int n_in,
                              void* d_out, int out_size, void* d_ws, size_t ws_size,
                              hipStream_t stream) {
    (void)in_sizes; (void)n_in; (void)out_size; (void)ws_size;
    const float* img1 = (const float*)d_in[0];
    const float* img2 = (const float*)d_in[1];
    float* partial = (float*)d_ws;               // NBLOCKS floats (48 KB)
    float* out = (float*)d_out;

    dim3 grid(NTILES, NTILES, NPLANES);
    ssim_tile_kernel<<<grid, 128, 0, stream>>>(img1, img2, partial);
    ssim_final_kernel<<<1, 256, 0, stream>>>(partial, out, NBLOCKS);
}